// ViTSelfAttention_6957847020291
// MI455X (gfx1250) — compile-verified
//
#include <hip/hip_runtime.h>
#include <stdint.h>

#define N_ROWS 32768
#define K_CLUST 1024
#define DIM 256
#define HEADS 8

typedef __attribute__((ext_vector_type(2))) float    v2f;
typedef __attribute__((ext_vector_type(8))) float    v8f;
typedef __attribute__((ext_vector_type(4))) uint32_t v4u;
typedef __attribute__((ext_vector_type(8))) int      v8i;
typedef __attribute__((ext_vector_type(4))) int      v4i;

__device__ __forceinline__ v8f wmma_f32(v2f a, v2f b, v8f c) {
  // V_WMMA_F32_16X16X4_F32 : D = A(16x4,f32) * B(4x16,f32) + C(16x16,f32)
  return __builtin_amdgcn_wmma_f32_16x16x4_f32(false, a, false, b, (short)0, c,
                                               false, false);
}

// ---------------------------------------------------------------------------
// K0: Wsum[r][c] = sum_h Wq[h*256 + r][c];  bsum[c] = sum_h bq[h*256 + c]
// ---------------------------------------------------------------------------
__global__ void k_reduce_wq(const float* __restrict__ Wq,
                            const float* __restrict__ bq,
                            float* __restrict__ Wsum,
                            float* __restrict__ bsum) {
  const int r = blockIdx.x;
  const int c = threadIdx.x;
  float s = 0.0f;
#pragma unroll
  for (int h = 0; h < HEADS; ++h) s += Wq[(size_t)(h * DIM + r) * DIM + c];
  Wsum[(size_t)r * DIM + c] = s;
  if (r == 0) {
    float b = 0.0f;
#pragma unroll
    for (int h = 0; h < HEADS; ++h) b += bq[h * DIM + c];
    bsum[c] = b;
  }
}

// ---------------------------------------------------------------------------
// K1/K4: C[m][j] = sum_d A[m][d] * B[j][d] + bias[j]   (C = A @ B^T + bias)
// One wave per 16(m) x 64(j) tile; f32 WMMA 16x16x4; grid = (Mrows/16)*4/8.
// ---------------------------------------------------------------------------
__global__ void k_gemm_at(const float* __restrict__ A,
                          const float* __restrict__ B,
                          const float* __restrict__ bias,
                          float* __restrict__ C) {
  const int wave = blockIdx.x * (blockDim.x >> 5) + (threadIdx.x >> 5);
  const int lane = threadIdx.x & 31;
  const int m0 = (wave >> 2) * 16;
  const int j0 = (wave & 3) * 64;
  const int lm = lane & 15;           // M (for A) / N (for B) index
  const int lk = (lane >> 4) * 2;     // K sub-offset {0,2}

  v8f acc0 = {}, acc1 = {}, acc2 = {}, acc3 = {};
  const float* arow = A + (size_t)(m0 + lm) * DIM + lk;
  const float* b0p = B + (size_t)(j0 + 0 + lm) * DIM + lk;
  const float* b1p = B + (size_t)(j0 + 16 + lm) * DIM + lk;
  const float* b2p = B + (size_t)(j0 + 32 + lm) * DIM + lk;
  const float* b3p = B + (size_t)(j0 + 48 + lm) * DIM + lk;

  for (int d = 0; d < DIM; d += 4) {
    v2f a  = *(const v2f*)(arow + d);
    v2f b0 = *(const v2f*)(b0p + d);
    v2f b1 = *(const v2f*)(b1p + d);
    v2f b2 = *(const v2f*)(b2p + d);
    v2f b3 = *(const v2f*)(b3p + d);
    acc0 = wmma_f32(a, b0, acc0);
    acc1 = wmma_f32(a, b1, acc1);
    acc2 = wmma_f32(a, b2, acc2);
    acc3 = wmma_f32(a, b3, acc3);
  }

  const int rbase = 8 * (lane >> 4);  // C/D layout: M = r + 8*(lane>>4), N = lm
#pragma unroll
  for (int r = 0; r < 8; ++r) {
    const size_t m = (size_t)(m0 + r + rbase);
    C[m * DIM + j0 +  0 + lm] = acc0[r] + bias[j0 +  0 + lm];
    C[m * DIM + j0 + 16 + lm] = acc1[r] + bias[j0 + 16 + lm];
    C[m * DIM + j0 + 32 + lm] = acc2[r] + bias[j0 + 32 + lm];
    C[m * DIM + j0 + 48 + lm] = acc3[r] + bias[j0 + 48 + lm];
  }
}

// ---------------------------------------------------------------------------
// K2: per 16-row block: stage qs tile to LDS via TDM, WMMA logits = qs@c^T/16,
//     row softmax -> p, idx = argmax(p + gumbel).
// ---------------------------------------------------------------------------
__launch_bounds__(512)
__global__ void k_logits_argmax(const float* __restrict__ qs,
                                const float* __restrict__ cmat,
                                const float* __restrict__ gum,
                                int* __restrict__ idx) {
  __shared__ float qsTile[16 * DIM];      // 16 KB
  __shared__ float lbuf[16 * K_CLUST];    // 64 KB
  const int n0 = blockIdx.x * 16;
  const int w = threadIdx.x >> 5;         // wave 0..15
  const int lane = threadIdx.x & 31;

  if (w == 0) {
    // Tensor DMA descriptor (D#) per CDNA5 ISA 8.3/8.4:
    // 2D tile: 256 f32 per row, 16 rows, row stride 256 elements.
    const uint64_t gaddr = (uint64_t)(uintptr_t)(qs + (size_t)n0 * DIM);
    const uint32_t ldsa = (uint32_t)(uintptr_t)qsTile;  // low 32 bits = LDS offset
    v4u g0;
    g0.x = 1u;                                           // count=1
    g0.y = ldsa;                                         // lds_addr
    g0.z = (uint32_t)gaddr;                              // global_addr[31:0]
    g0.w = (uint32_t)((gaddr >> 32) & 0x01FFFFFFu) | (2u << 30);  // addr[56:32], type=2
    v8i g1;
    g1[0] = (int)(2u << 16);     // data_size = 2 (4 bytes), no mask/pad/iterate
    g1[1] = (int)(256u << 16);   // tensor_dim0 = 256 (low 16)
    g1[2] = (int)(16u << 16);    // tensor_dim0 hi=0 | tensor_dim1 = 16 (low 16)
    g1[3] = (int)(256u << 16);   // tensor_dim1 hi=0 | tile_dim0 = 256
    g1[4] = 16;                  // tile_dim1 = 16 | tile_dim2 = 0
    g1[5] = 256;                 // tensor_dim0_stride low 32 = 256
    g1[6] = 0;                   // stride hi | tensor_dim1_stride lo
    g1[7] = 0;
    v4i gz = {};
#if __clang_major__ >= 23
    v8i gz8 = {};
    __builtin_amdgcn_tensor_load_to_lds(g0, g1, gz, gz, gz8, 0);
#else
    __builtin_amdgcn_tensor_load_to_lds(g0, g1, gz, gz, 0);
#endif
    __builtin_amdgcn_s_wait_tensorcnt(0);
  }
  __syncthreads();

  // Each wave: 16 rows x 64 cols of logits, cols k0 = w*64.
  const int k0 = w * 64;
  const int lm = lane & 15;
  const int lk = (lane >> 4) * 2;
  v8f acc0 = {}, acc1 = {}, acc2 = {}, acc3 = {};
  const float* a_lds = qsTile + lm * DIM + lk;
  const float* c0p = cmat + (size_t)(k0 + 0 + lm) * DIM + lk;
  const float* c1p = cmat + (size_t)(k0 + 16 + lm) * DIM + lk;
  const float* c2p = cmat + (size_t)(k0 + 32 + lm) * DIM + lk;
  const float* c3p = cmat + (size_t)(k0 + 48 + lm) * DIM + lk;
  for (int d = 0; d < DIM; d += 4) {
    v2f a  = *(const v2f*)(a_lds + d);     // ds_load_b64
    v2f b0 = *(const v2f*)(c0p + d);
    v2f b1 = *(const v2f*)(c1p + d);
    v2f b2 = *(const v2f*)(c2p + d);
    v2f b3 = *(const v2f*)(c3p + d);
    acc0 = wmma_f32(a, b0, acc0);
    acc1 = wmma_f32(a, b1, acc1);
    acc2 = wmma_f32(a, b2, acc2);
    acc3 = wmma_f32(a, b3, acc3);
  }
  const float scal = 0.0625f;  // 1/sqrt(256)
  const int rbase = 8 * (lane >> 4);
#pragma unroll
  for (int r = 0; r < 8; ++r) {
    const int row = r + rbase;
    lbuf[row * K_CLUST + k0 +  0 + lm] = acc0[r] * scal;
    lbuf[row * K_CLUST + k0 + 16 + lm] = acc1[r] * scal;
    lbuf[row * K_CLUST + k0 + 32 + lm] = acc2[r] * scal;
    lbuf[row * K_CLUST + k0 + 48 + lm] = acc3[r] * scal;
  }
  __syncthreads();

  // Wave w handles row w: softmax over 1024 cols, then argmax(p + g).
  float* lrow = lbuf + w * K_CLUST;
  float mx = -__builtin_inff();
  for (int k = lane; k < K_CLUST; k += 32) mx = fmaxf(mx, lrow[k]);
#pragma unroll
  for (int off = 16; off; off >>= 1) mx = fmaxf(mx, __shfl_xor(mx, off, 32));
  float s = 0.0f;
  for (int k = lane; k < K_CLUST; k += 32) {
    const float e = __expf(lrow[k] - mx);
    lrow[k] = e;                       // same lane re-reads its own entries
    s += e;
  }
#pragma unroll
  for (int off = 16; off; off >>= 1) s += __shfl_xor(s, off, 32);
  const float inv = 1.0f / s;
  float bestV = -__builtin_inff();
  int bestI = 0x7fffffff;
  const float* grow = gum + (size_t)(n0 + w) * K_CLUST;
  for (int k = lane; k < K_CLUST; k += 32) {
    const float v = lrow[k] * inv + grow[k];
    if (v > bestV) { bestV = v; bestI = k; }
  }
#pragma unroll
  for (int off = 16; off; off >>= 1) {
    const float ov = __shfl_xor(bestV, off, 32);
    const int oi = __shfl_xor(bestI, off, 32);
    if (ov > bestV || (ov == bestV && oi < bestI)) { bestV = ov; bestI = oi; }
  }
  if (lane == 0) idx[n0 + w] = bestI;
}

// ---------------------------------------------------------------------------
// K3: accum[idx[n]][:] += x1[n][:]  via global_atomic_add_f32
// ---------------------------------------------------------------------------
__global__ void k_scatter(const float* __restrict__ x1,
                          const int* __restrict__ idx,
                          float* __restrict__ accum) {
  const int n = blockIdx.x;
  const int d = threadIdx.x;
  const int k = idx[n];
  unsafeAtomicAdd(&accum[(size_t)k * DIM + d], x1[(size_t)n * DIM + d]);
}

// ---------------------------------------------------------------------------
extern "C" void kernel_launch(void* const* d_in, const int* in_sizes, int n_in,
                              void* d_out, int out_size, void* d_ws, size_t ws_size,
                              hipStream_t stream) {
  const float* x1  = (const float*)d_in[0];
  const float* c   = (const float*)d_in[1];
  const float* Wq  = (const float*)d_in[2];
  const float* bq  = (const float*)d_in[3];
  const float* Wd  = (const float*)d_in[4];
  const float* bd  = (const float*)d_in[5];
  const float* gum = (const float*)d_in[6];
  float* out = (float*)d_out;

  char* p = (char*)d_ws;
  float* Wsum  = (float*)p; p += (size_t)DIM * DIM * sizeof(float);
  float* bsum  = (float*)p; p += DIM * sizeof(float);
  float* qs    = (float*)p; p += (size_t)N_ROWS * DIM * sizeof(float);
  int*   idx   = (int*)p;   p += (size_t)N_ROWS * sizeof(int);
  float* accum = (float*)p; p += (size_t)K_CLUST * DIM * sizeof(float);

  // out[0 .. N*D) = x1 (passthrough output #1)
  hipMemcpyAsync(out, x1, (size_t)N_ROWS * DIM * sizeof(float),
                 hipMemcpyDeviceToDevice, stream);
  hipMemsetAsync(accum, 0, (size_t)K_CLUST * DIM * sizeof(float), stream);

  // Wsum/bsum
  k_reduce_wq<<<DIM, DIM, 0, stream>>>(Wq, bq, Wsum, bsum);
  // qs = x1 @ Wsum^T + bsum : waves = (32768/16)*4 = 8192 -> 1024 blocks x 8 waves
  k_gemm_at<<<1024, 256, 0, stream>>>(x1, Wsum, bsum, qs);
  // fused logits/softmax/argmax
  k_logits_argmax<<<N_ROWS / 16, 512, 0, stream>>>(qs, c, gum, idx);
  // scatter-sum of x1 rows by idx
  k_scatter<<<N_ROWS, DIM, 0, stream>>>(x1, idx, accum);
  // out2 = accum @ Wd^T + bd : waves = (1024/16)*4 = 256 -> 32 blocks x 8 waves
  k_gemm_at<<<32, 256, 0, stream>>>(accum, Wd, bd, out + (size_t)N_ROWS * DIM);
}